// CombinedLossExp72_18064632446984
// MI455X (gfx1250) — compile-verified
//
#include <hip/hip_runtime.h>
#include <math.h>

// ---------------------------------------------------------------------------
// Combined distillation loss for MI455X (gfx1250, wave32, WMMA).
//   B=8 T=2048 D=512 K=4096  N=B*T=16384
// Heavy stage: dot = x.c via V_WMMA_F32_16X16X4_F32 (4 column tiles per wave,
// one A fragment feeds 4 WMMAs), score = fmaf(-2, dot, c2), fused per-row
// top-16 selection (never materializes the 256MB d2 matrix).
// X tile staged to LDS with GLOBAL_LOAD_ASYNC_TO_LDS_B128 (ASYNCcnt path).
// ---------------------------------------------------------------------------

typedef float v2f __attribute__((ext_vector_type(2)));
typedef float v4f __attribute__((ext_vector_type(4)));
typedef float v8f __attribute__((ext_vector_type(8)));

#define B_ 8
#define T_ 2048
#define D_ 512
#define K_ 4096
#define N_ (B_ * T_)
#define NNEG 16
#define MARGIN_ 0.2f
#define INV_TEMP 10.0f
#define WPB 4   // waves per block in the GEMM kernel
#define NT 4    // 16-wide column tiles processed concurrently per wave

__device__ __forceinline__ float wave_sum(float v) {
#pragma unroll
  for (int o = 16; o > 0; o >>= 1) v += __shfl_xor(v, o, 32);
  return v;
}
__device__ __forceinline__ float wave_max(float v) {
#pragma unroll
  for (int o = 16; o > 0; o >>= 1) v = fmaxf(v, __shfl_xor(v, o, 32));
  return v;
}
__device__ __forceinline__ float dot4(v4f a, v4f b) {
  return a.x * b.x + a.y * b.y + a.z * b.z + a.w * b.w;
}

// ---------------------------------------------------------------------------
// K0: zero the 3 loss accumulators.
// ---------------------------------------------------------------------------
__global__ void k_zero(float* __restrict__ accum) {
  if (threadIdx.x < 3) accum[threadIdx.x] = 0.0f;
}

// ---------------------------------------------------------------------------
// K1: c2[k] = ||codebook[k]||^2.  One wave per codebook row.
// ---------------------------------------------------------------------------
__global__ __launch_bounds__(256) void k_c2(const float* __restrict__ C,
                                            float* __restrict__ c2) {
  int lane = threadIdx.x & 31;
  int row = blockIdx.x * 8 + (threadIdx.x >> 5);
  const v4f* c4 = (const v4f*)(C + (size_t)row * D_);
  float s = 0.0f;
  for (int j = lane; j < D_ / 4; j += 32) {
    v4f v = c4[j];
    s += dot4(v, v);
  }
  s = wave_sum(s);
  if (lane == 0) c2[row] = s;
}

// ---------------------------------------------------------------------------
// K2: feature (masked-MSE) + triplet losses.  One wave per token.
// neg_teacher = roll(teacher, shift=1, axis=0).
// ---------------------------------------------------------------------------
__global__ __launch_bounds__(256) void k_feat_trip(const float* __restrict__ S,
                                                   const float* __restrict__ T,
                                                   float* __restrict__ accum) {
  int lane = threadIdx.x & 31;
  int n = blockIdx.x * 8 + (threadIdx.x >> 5);
  int b = n >> 11, t = n & (T_ - 1);
  int nneg = (((b + B_ - 1) & (B_ - 1)) << 11) + t;
  const v4f* s4 = (const v4f*)(S + (size_t)n * D_);
  const v4f* t4 = (const v4f*)(T + (size_t)n * D_);
  const v4f* g4 = (const v4f*)(T + (size_t)nneg * D_);
  float pos = 0.0f, neg = 0.0f;
  for (int j = lane; j < D_ / 4; j += 32) {
    v4f sv = s4[j];
    v4f dp = sv - t4[j];
    v4f dn = sv - g4[j];
    pos += dot4(dp, dp);
    neg += dot4(dn, dn);
  }
  pos = wave_sum(pos);
  neg = wave_sum(neg);
  if (lane == 0) {
    atomicAdd(&accum[0], pos * (1.0f / D_));                        // feature
    float tl = sqrtf(pos) - sqrtf(neg) + MARGIN_;
    atomicAdd(&accum[1], fmaxf(tl, 0.0f));                         // triplet
  }
}

// ---------------------------------------------------------------------------
// K3: fused fp32 WMMA GEMM + per-row top-16 smallest distance.
// One wave owns a 16-row tile; walks K in groups of NT*16=64 columns so each
// A fragment (LDS) feeds NT WMMAs.  X tile staged via async global->LDS DMA.
// ---------------------------------------------------------------------------
__global__ __launch_bounds__(WPB * 32) void k_gemm_topk(
    const float* __restrict__ X, const float* __restrict__ C,
    const float* __restrict__ c2, const int* __restrict__ codes,
    int* __restrict__ negIdx) {
  __shared__ float As[WPB][16][D_];        // X tile, 32KB per wave
  __shared__ float St[WPB][16][NT * 16];   // score tile scratch

  const int lane = threadIdx.x & 31;
  const int w = threadIdx.x >> 5;
  const int rowbase = (blockIdx.x * WPB + w) * 16;

  // ---- Stage X tile (16 rows x 512 = 2048 float4s) via async DMA to LDS.
  {
    unsigned long long gbase =
        (unsigned long long)(const void*)(X + (size_t)rowbase * D_);
    unsigned ldsbase = (unsigned)(size_t)(void*)&As[w][0][0];
    for (int i = lane; i < 16 * (D_ / 4); i += 32) {
      unsigned laddr = ldsbase + (unsigned)i * 16u;
      unsigned long long gaddr = gbase + (unsigned long long)i * 16ull;
      asm volatile("global_load_async_to_lds_b128 %0, %1, off"
                   :: "v"(laddr), "v"(gaddr)
                   : "memory");
    }
    asm volatile("s_wait_asynccnt 0x0" ::: "memory");
  }
  __syncthreads();

  // WMMA f32 16x16x4 fragment addressing (wave32):
  //  A 16x4: lanes 0-15 -> M=lane, K={0,1}; lanes 16-31 -> M=lane-16, K={2,3}
  //  B 4x16: symmetric with N instead of M.  C/D: lane = col, 8 rows/lane.
  const int m = lane & 15;
  const int koff = (lane < 16) ? 0 : 2;
  const int r0 = (lane < 16) ? 0 : 8;

  float tv[NNEG];
  int ti[NNEG];
#pragma unroll
  for (int j = 0; j < NNEG; ++j) {
    tv[j] = __builtin_huge_valf();
    ti[j] = 0;
  }
  const int mycode = (lane < 16) ? codes[rowbase + lane] : -1;

  const float* ap = &As[w][m][koff];

  for (int colbase = 0; colbase < K_; colbase += NT * 16) {
    v8f acc0 = {}, acc1 = {}, acc2 = {}, acc3 = {};
    const float* bp = C + (size_t)(colbase + m) * D_ + koff;
#pragma unroll 2
    for (int d0 = 0; d0 < D_; d0 += 4) {
      v2f a  = *(const v2f*)(ap + d0);                 // ds_load_b64 (shared)
      v2f b0 = *(const v2f*)(bp + d0);                 // codebook: L2-resident
      v2f b1 = *(const v2f*)(bp + 16 * D_ + d0);
      v2f b2 = *(const v2f*)(bp + 32 * D_ + d0);
      v2f b3 = *(const v2f*)(bp + 48 * D_ + d0);
      acc0 = __builtin_amdgcn_wmma_f32_16x16x4_f32(false, a, false, b0,
                                                   (short)0, acc0, false, false);
      acc1 = __builtin_amdgcn_wmma_f32_16x16x4_f32(false, a, false, b1,
                                                   (short)0, acc1, false, false);
      acc2 = __builtin_amdgcn_wmma_f32_16x16x4_f32(false, a, false, b2,
                                                   (short)0, acc2, false, false);
      acc3 = __builtin_amdgcn_wmma_f32_16x16x4_f32(false, a, false, b3,
                                                   (short)0, acc3, false, false);
    }
    // score = c2[col] - 2*dot  (ranking-equivalent to d2: x2 is row-constant)
    {
      float cA = c2[colbase + m];
      float cB = c2[colbase + 16 + m];
      float cC = c2[colbase + 32 + m];
      float cD = c2[colbase + 48 + m];
#pragma unroll
      for (int r = 0; r < 8; ++r) {
        St[w][r0 + r][m]      = fmaf(-2.0f, acc0[r], cA);
        St[w][r0 + r][16 + m] = fmaf(-2.0f, acc1[r], cB);
        St[w][r0 + r][32 + m] = fmaf(-2.0f, acc2[r], cC);
        St[w][r0 + r][48 + m] = fmaf(-2.0f, acc3[r], cD);
      }
    }
    __builtin_amdgcn_wave_barrier();

    // Lanes 0-15: one row each; keep 16 smallest scores (sorted ascending).
    if (lane < 16) {
#pragma unroll 1
      for (int c = 0; c < NT * 16; ++c) {
        int idx = colbase + c;
        float s = St[w][lane][c];
        if (idx == mycode) continue;     // mask positive (reference: +inf)
        if (s < tv[NNEG - 1]) {          // early-out: beats current worst?
          float cs = s;
          int ci = idx;
#pragma unroll
          for (int j = 0; j < NNEG; ++j) {
            if (cs < tv[j]) {
              float tf = tv[j]; int tx = ti[j];
              tv[j] = cs;  ti[j] = ci;
              cs = tf;     ci = tx;
            }
          }
        }
      }
    }
    __builtin_amdgcn_wave_barrier();
  }

  if (lane < 16) {
#pragma unroll
    for (int j = 0; j < NNEG; ++j)
      negIdx[(size_t)(rowbase + lane) * NNEG + j] = ti[j];
  }
}

// ---------------------------------------------------------------------------
// K4: InfoNCE over [positive, 16 mined negatives].  One wave per token;
// lane j in [0,16] owns one candidate vector (codebook rows are L2-resident).
// ---------------------------------------------------------------------------
__global__ __launch_bounds__(256) void k_contrastive(
    const float* __restrict__ X, const float* __restrict__ C,
    const int* __restrict__ codes, const int* __restrict__ negIdx,
    float* __restrict__ accum) {
  int lane = threadIdx.x & 31;
  int n = blockIdx.x * 8 + (threadIdx.x >> 5);
  const v4f* x4 = (const v4f*)(X + (size_t)n * D_);

  // ||x||
  float s2 = 0.0f;
  for (int j = lane; j < D_ / 4; j += 32) {
    v4f v = x4[j];
    s2 += dot4(v, v);
  }
  s2 = wave_sum(s2);
  float invx = 1.0f / fmaxf(sqrtf(s2), 1e-12f);

  // lane 0 = positive, lanes 1..16 = negatives, lanes 17..31 dummy (masked).
  bool valid = (lane <= NNEG);
  int vrow = (lane == 0 || !valid) ? codes[n]
                                   : negIdx[(size_t)n * NNEG + lane - 1];
  const v4f* c4 = (const v4f*)(C + (size_t)vrow * D_);
  float dotv = 0.0f, v2 = 0.0f;
#pragma unroll 4
  for (int j = 0; j < D_ / 4; ++j) {
    v4f xv = x4[j], cv = c4[j];
    dotv += dot4(xv, cv);
    v2 += dot4(cv, cv);
  }
  float invv = 1.0f / fmaxf(sqrtf(v2), 1e-12f);
  float logit = dotv * invx * invv * INV_TEMP;

  float mx = wave_max(valid ? logit : -__builtin_huge_valf());
  float se = wave_sum(valid ? expf(logit - mx) : 0.0f);
  float l0 = __shfl(logit, 0, 32);
  if (lane == 0) atomicAdd(&accum[2], (mx + logf(se)) - l0);  // -log_softmax[0]
}

// ---------------------------------------------------------------------------
// K5: total = feature/N + triplet/N + 0.5 * contrastive/N
// ---------------------------------------------------------------------------
__global__ void k_finalize(const float* __restrict__ accum,
                           float* __restrict__ out) {
  out[0] = (accum[0] + accum[1] + 0.5f * accum[2]) * (1.0f / N_);
}

// ---------------------------------------------------------------------------
extern "C" void kernel_launch(void* const* d_in, const int* in_sizes, int n_in,
                              void* d_out, int out_size, void* d_ws,
                              size_t ws_size, hipStream_t stream) {
  const float* S = (const float*)d_in[0];      // student (B,T,D)
  const float* T = (const float*)d_in[1];      // teacher (B,T,D)
  const float* C = (const float*)d_in[2];      // codebook (K,D)
  const int* codes = (const int*)d_in[3];      // teacher codes (B,T)
  float* out = (float*)d_out;

  // workspace: [0..2] accumulators | [16..16+K) c2 | N*NNEG int neg indices
  float* accum = (float*)d_ws;
  float* c2 = accum + 16;
  int* negIdx = (int*)(c2 + K_);

  k_zero<<<1, 32, 0, stream>>>(accum);
  k_c2<<<K_ / 8, 256, 0, stream>>>(C, c2);
  k_feat_trip<<<N_ / 8, 256, 0, stream>>>(S, T, accum);
  k_gemm_topk<<<N_ / (16 * WPB), WPB * 32, 0, stream>>>(S, C, c2, codes,
                                                        negIdx);
  k_contrastive<<<N_ / 8, 256, 0, stream>>>(S, C, codes, negIdx, accum);
  k_finalize<<<1, 1, 0, stream>>>(accum, out);
}